// AdaptiveRankSSM_74053826118136
// MI455X (gfx1250) — compile-verified
//
#include <hip/hip_runtime.h>
#include <math.h>

// ---------------- problem constants ----------------
#define D_MODEL 1024
#define D_STATE 64
#define MAX_RANK 32
#define BATCH 8
#define SEQ 4096
#define NROWS (BATCH * SEQ)          // 32768
#define CHUNK 128                    // scan chunk length (= 2^7)
#define NCHUNK (SEQ / CHUNK)         // 32
#define N_OUT (NROWS * D_MODEL)      // 33554432 floats in d_out before scalars

typedef __attribute__((ext_vector_type(2))) float v2f;
typedef __attribute__((ext_vector_type(8))) float v8f;

// =====================================================================
// 1) A = A_low @ A_high   (64x64, rank 32)
// =====================================================================
__global__ __launch_bounds__(256) void k_compute_A(const float* __restrict__ Alow,
                                                   const float* __restrict__ Ahigh,
                                                   float* __restrict__ A) {
  int tid = threadIdx.x;
  for (int i = tid; i < D_STATE * D_STATE; i += 256) {
    int n = i >> 6, m = i & 63;
    float s = 0.f;
#pragma unroll
    for (int r = 0; r < MAX_RANK; ++r) s += Alow[n * MAX_RANK + r] * Ahigh[r * D_STATE + m];
    A[i] = s;
  }
}

// =====================================================================
// 2) AL = A^CHUNK via 7 repeated squarings (CHUNK = 128 = 2^7)
// =====================================================================
__global__ __launch_bounds__(256) void k_matpow(const float* __restrict__ A,
                                                float* __restrict__ AL) {
  __shared__ float M[4096];
  __shared__ float N[4096];
  int tid = threadIdx.x;
  for (int i = tid; i < 4096; i += 256) M[i] = A[i];
  __syncthreads();
  for (int s = 0; s < 7; ++s) {
    for (int i = tid; i < 4096; i += 256) {
      int n = i >> 6, m = i & 63;
      float acc = 0.f;
#pragma unroll 8
      for (int k = 0; k < 64; ++k) acc += M[n * 64 + k] * M[k * 64 + m];
      N[i] = acc;
    }
    __syncthreads();
    for (int i = tid; i < 4096; i += 256) M[i] = N[i];
    __syncthreads();
  }
  for (int i = tid; i < 4096; i += 256) AL[i] = M[i];
}

// =====================================================================
// 3) spectral norm of A via power iteration on B = A^T A.
//    After convergence ||Bv|| = sigma^2, so sigma = ss^(1/4).
// =====================================================================
__global__ __launch_bounds__(64) void k_spectral(const float* __restrict__ A,
                                                 float* __restrict__ outp) {
  __shared__ float As[64 * 65];  // padded rows: bank-conflict free both by row and column
  __shared__ float vv[64], ww[64], uu[64];
  int tid = threadIdx.x;
  for (int i = tid; i < 4096; i += 64) As[(i >> 6) * 65 + (i & 63)] = A[i];
  vv[tid] = 0.125f;  // unit norm start
  __syncthreads();
  float ss = 0.f;
  for (int it = 0; it < 128; ++it) {
    float w = 0.f;
#pragma unroll 8
    for (int m = 0; m < 64; ++m) w += As[tid * 65 + m] * vv[m];   // w = A v
    ww[tid] = w;
    __syncthreads();
    float u = 0.f;
#pragma unroll 8
    for (int n = 0; n < 64; ++n) u += As[n * 65 + tid] * ww[n];   // u = A^T w
    uu[tid] = u;
    __syncthreads();
    ss = 0.f;
#pragma unroll 8
    for (int i = 0; i < 64; ++i) ss += uu[i] * uu[i];             // ||Bv||^2 (broadcast reads)
    float inv = rsqrtf(ss > 0.f ? ss : 1.f);
    vv[tid] = u * inv;
    __syncthreads();
  }
  if (tid == 0) outp[0] = powf(ss, 0.25f);
}

// =====================================================================
// 4) u = (x @ B_w^T + B_b) * rank_weights   — WMMA f32 GEMM
//    M = 32768, N = 64, K = 1024. Block = 8 waves = 32 rows x 64 cols.
// =====================================================================
__global__ __launch_bounds__(256) void k_u_gemm(const float* __restrict__ x,
                                                const float* __restrict__ Bw,
                                                const float* __restrict__ Bb,
                                                const float* __restrict__ rw,
                                                float* __restrict__ u) {
  int tid = threadIdx.x;
  int lane = tid & 31, w = tid >> 5;
  int mtile = w >> 2, ntile = w & 3;
  int row0 = blockIdx.x * 32 + mtile * 16;
  int col0 = ntile * 16;
  int lrow = lane & 15;            // A row / B col within tile
  int koff = (lane >> 4) << 1;     // lanes 16-31 hold K=2,3 of each 4-step
  const float* xp = x + (size_t)(row0 + lrow) * D_MODEL + koff;
  const float* bp = Bw + (size_t)(col0 + lrow) * D_MODEL + koff;  // B(k,n) = B_w[n,k], contiguous in k
  v8f acc = {};
#pragma unroll 8
  for (int k = 0; k < D_MODEL; k += 4) {
    v2f a = *(const v2f*)(xp + k);
    v2f b = *(const v2f*)(bp + k);
    acc = __builtin_amdgcn_wmma_f32_16x16x4_f32(false, a, false, b, (short)0, acc, false, false);
  }
  int lhi = (lane >> 4) * 8;
  int lcol = lane & 15;
#pragma unroll
  for (int v = 0; v < 8; ++v) {
    int r = row0 + v + lhi;
    int c = col0 + lcol;
    u[(size_t)r * D_STATE + c] = (acc[v] + Bb[c]) * rw[r];
  }
}

// =====================================================================
// 5) per-chunk local scan: g_t = A g_{t-1} + u_t (g=0 at chunk start),
//    written in place over u; chunk-end state -> E.  64 threads/block.
// =====================================================================
__global__ __launch_bounds__(64) void k_local_scan(float* __restrict__ hs,
                                                   const float* __restrict__ A,
                                                   float* __restrict__ E) {
  __shared__ float As[64 * 65];
  __shared__ float hb[2][64];
  int tid = threadIdx.x;
  int b = blockIdx.x / NCHUNK, c = blockIdx.x % NCHUNK;
  for (int i = tid; i < 4096; i += 64) As[(i >> 6) * 65 + (i & 63)] = A[i];
  hb[0][tid] = 0.f;
  __syncthreads();
  int base = (b * SEQ + c * CHUNK) * D_STATE;
  int cur = 0;
  for (int j = 0; j < CHUNK; ++j) {
    float acc = hs[base + j * D_STATE + tid];
    const float* ar = &As[tid * 65];
#pragma unroll
    for (int m = 0; m < 64; ++m) acc += ar[m] * hb[cur][m];
    hs[base + j * D_STATE + tid] = acc;
    hb[cur ^ 1][tid] = acc;
    cur ^= 1;
    __syncthreads();
  }
  E[(b * NCHUNK + c) * D_STATE + tid] = hb[cur][tid];
}

// =====================================================================
// 6) serial chunk combine (per batch): carry[c+1] = A^L carry[c] + E[c]
// =====================================================================
__global__ __launch_bounds__(64) void k_combine(const float* __restrict__ AL,
                                                const float* __restrict__ E,
                                                float* __restrict__ Cy) {
  __shared__ float As[64 * 65];
  __shared__ float hb[2][64];
  int tid = threadIdx.x, b = blockIdx.x;
  for (int i = tid; i < 4096; i += 64) As[(i >> 6) * 65 + (i & 63)] = AL[i];
  hb[0][tid] = 0.f;
  __syncthreads();
  int cur = 0;
  for (int c = 0; c < NCHUNK; ++c) {
    Cy[(b * NCHUNK + c) * D_STATE + tid] = hb[cur][tid];  // carry INTO chunk c
    float acc = E[(b * NCHUNK + c) * D_STATE + tid];
    const float* ar = &As[tid * 65];
#pragma unroll
    for (int m = 0; m < 64; ++m) acc += ar[m] * hb[cur][m];
    hb[cur ^ 1][tid] = acc;
    cur ^= 1;
    __syncthreads();
  }
}

// =====================================================================
// 7) correction pass: p <- A p (p0 = carry into chunk); hs[t] += p
// =====================================================================
__global__ __launch_bounds__(64) void k_correct(float* __restrict__ hs,
                                                const float* __restrict__ A,
                                                const float* __restrict__ Cy) {
  int tid = threadIdx.x;
  int b = blockIdx.x / NCHUNK, c = blockIdx.x % NCHUNK;
  if (c == 0) return;  // carry into chunk 0 is zero
  __shared__ float As[64 * 65];
  __shared__ float hb[2][64];
  for (int i = tid; i < 4096; i += 64) As[(i >> 6) * 65 + (i & 63)] = A[i];
  hb[0][tid] = Cy[(b * NCHUNK + c) * D_STATE + tid];
  __syncthreads();
  int base = (b * SEQ + c * CHUNK) * D_STATE;
  int cur = 0;
  for (int j = 0; j < CHUNK; ++j) {
    float acc = 0.f;
    const float* ar = &As[tid * 65];
#pragma unroll
    for (int m = 0; m < 64; ++m) acc += ar[m] * hb[cur][m];
    hs[base + j * D_STATE + tid] += acc;
    hb[cur ^ 1][tid] = acc;
    cur ^= 1;
    __syncthreads();
  }
}

// =====================================================================
// 8) out = hs @ C_w^T + C_b + D*x   — WMMA f32 GEMM, K = 64
//    Block = 8 waves = 16 rows x 128 cols; grid (2048, 8).
// =====================================================================
__global__ __launch_bounds__(256) void k_out_gemm(const float* __restrict__ hs,
                                                  const float* __restrict__ Cw,
                                                  const float* __restrict__ Cb,
                                                  const float* __restrict__ Dv,
                                                  const float* __restrict__ x,
                                                  float* __restrict__ out) {
  int tid = threadIdx.x;
  int lane = tid & 31, w = tid >> 5;
  int row0 = blockIdx.x * 16;
  int col0 = blockIdx.y * 128 + w * 16;
  int lrow = lane & 15;
  int koff = (lane >> 4) << 1;
  const float* ap = hs + (size_t)(row0 + lrow) * D_STATE + koff;
  const float* bp = Cw + (size_t)(col0 + lrow) * D_STATE + koff;  // B(k,d) = C_w[d,k], contiguous in k
  v8f acc = {};
#pragma unroll
  for (int k = 0; k < D_STATE; k += 4) {
    v2f a = *(const v2f*)(ap + k);
    v2f b = *(const v2f*)(bp + k);
    acc = __builtin_amdgcn_wmma_f32_16x16x4_f32(false, a, false, b, (short)0, acc, false, false);
  }
  int lhi = (lane >> 4) * 8;
  int lcol = lane & 15;
#pragma unroll
  for (int v = 0; v < 8; ++v) {
    int r = row0 + v + lhi;
    int c = col0 + lcol;
    out[(size_t)r * D_MODEL + c] = acc[v] + Cb[c] + Dv[c] * x[(size_t)r * D_MODEL + c];
  }
}

// =====================================================================
// 9) state_norm_mean = mean_b ||h_last[b]||
// =====================================================================
__global__ __launch_bounds__(64) void k_state_norm(const float* __restrict__ hs,
                                                   float* __restrict__ outp) {
  __shared__ float norms[BATCH];
  int tid = threadIdx.x;
  if (tid < BATCH) {
    size_t base = ((size_t)tid * SEQ + (SEQ - 1)) * D_STATE;
    float ss = 0.f;
    for (int i = 0; i < D_STATE; ++i) { float v = hs[base + i]; ss += v * v; }
    norms[tid] = sqrtf(ss);
  }
  __syncthreads();
  if (tid == 0) {
    float s = 0.f;
    for (int b = 0; b < BATCH; ++b) s += norms[b];
    outp[0] = s / (float)BATCH;
  }
}

// =====================================================================
extern "C" void kernel_launch(void* const* d_in, const int* in_sizes, int n_in,
                              void* d_out, int out_size, void* d_ws, size_t ws_size,
                              hipStream_t stream) {
  const float* x     = (const float*)d_in[0];
  const float* rw    = (const float*)d_in[1];
  const float* A_low = (const float*)d_in[2];
  const float* A_high= (const float*)d_in[3];
  const float* B_w   = (const float*)d_in[4];
  const float* B_b   = (const float*)d_in[5];
  const float* C_w   = (const float*)d_in[6];
  const float* C_b   = (const float*)d_in[7];
  const float* Dv    = (const float*)d_in[8];
  float* out = (float*)d_out;

  // workspace layout (floats)
  float* ws   = (float*)d_ws;
  float* wsA  = ws;            // 4096   : A (64x64)
  float* wsAL = ws + 4096;     // 4096   : A^128
  float* wsE  = ws + 8192;     // 16384  : chunk-end local states
  float* wsC  = ws + 24576;    // 16384  : carry into each chunk
  float* wsH  = ws + 40960;    // 2097152: u, overwritten by hs (in place)

  k_compute_A<<<1, 256, 0, stream>>>(A_low, A_high, wsA);
  k_matpow   <<<1, 256, 0, stream>>>(wsA, wsAL);
  k_spectral <<<1, 64, 0, stream>>>(wsA, out + N_OUT + 1);
  k_u_gemm   <<<NROWS / 32, 256, 0, stream>>>(x, B_w, B_b, rw, wsH);
  k_local_scan<<<BATCH * NCHUNK, 64, 0, stream>>>(wsH, wsA, wsE);
  k_combine  <<<BATCH, 64, 0, stream>>>(wsAL, wsE, wsC);
  k_correct  <<<BATCH * NCHUNK, 64, 0, stream>>>(wsH, wsA, wsC);
  dim3 og(NROWS / 16, D_MODEL / 128);
  k_out_gemm <<<og, 256, 0, stream>>>(wsH, C_w, C_b, Dv, x, out);
  k_state_norm<<<1, 64, 0, stream>>>(wsH, out + N_OUT);
}